// CgpHmmCell_70686571757886
// MI455X (gfx1250) — compile-verified
//
#include <hip/hip_runtime.h>
#include <hip/hip_bf16.h>

// Problem constants (match reference)
constexpr int HB = 512;    // batch
constexpr int HT = 4096;   // seq len
constexpr int HS = 64;     // states
constexpr int HM = 125;    // emission alphabet

typedef float v2f __attribute__((ext_vector_type(2)));
typedef float v8f __attribute__((ext_vector_type(8)));

// ---------------------------------------------------------------------------
// Kernel 1: one-hot [B*T, 125] -> int index. One wave32 per row.
// idx = sum_i i * x[i] is exact for a fp32 one-hot row.
// ---------------------------------------------------------------------------
__global__ __launch_bounds__(256) void obs_from_onehot(
    const float* __restrict__ x, int* __restrict__ obs, int nrows) {
  int gid  = blockIdx.x * 256 + threadIdx.x;
  int row  = gid >> 5;
  int lane = gid & 31;
  if (row >= nrows) return;
  const float* xr = x + (size_t)row * HM;
  float s = 0.f;
  int i0 = lane * 4;
#pragma unroll
  for (int j = 0; j < 4; ++j) {
    int i = i0 + j;
    if (i < HM) s += xr[i] * (float)i;
  }
#pragma unroll
  for (int off = 16; off > 0; off >>= 1) s += __shfl_xor(s, off, 32);
  if (lane == 0) obs[row] = (int)(s + 0.5f);
}

// ---------------------------------------------------------------------------
// Kernel 2: HMM forward scan. One wave32 per 16-batch tile (32 blocks).
//
// Per step:  R = alpha @ A  via 64x V_WMMA_F32_16X16X4_F32 (full fp32),
// then D-layout -> A-operand-layout transpose through LDS (stride 19,
// bank-conflict-free), fused with emission gather Bm[obs], row-sum Z via
// lane sum + shfl_xor(16), rescale, ll += logf(Z).
//
// Layouts (wave32, 32-bit WMMA operands), lane l: lo=l&15, hi=l>>4:
//   A-operand 16x4 chunk kc : vgpr v holds alpha[lo][4kc + 2*hi + v]
//   B-operand 4x16 (kc,nt)  : vgpr v holds A[4kc + 2*hi + v][16nt + lo]
//   C/D 16x16 tile nt       : vgpr r holds R[r + 8*hi][16nt + lo]
// ---------------------------------------------------------------------------
__global__ __launch_bounds__(32) void hmm_forward_scan(
    const int* __restrict__ obs, const float* __restrict__ I,
    const float* __restrict__ A, const float* __restrict__ Bm,
    float* __restrict__ out) {
  constexpr int LSTR = 19;               // LDS column stride (floats), conflict-free
  __shared__ float lds[HS * LSTR];       // R tile, element [row m][col s] at s*LSTR + m

  const int lane = threadIdx.x;
  const int lo = lane & 15;
  const int hi = lane >> 4;
  const int m  = blockIdx.x * 16 + lo;   // batch row owned by this lane (A-frag space)
  const int* obs_row = obs + (size_t)m * HT;

  // ---- Transition matrix as resident B-operand fragments (loaded once) ----
  v2f bf[16][4];
#pragma unroll
  for (int kc = 0; kc < 16; ++kc) {
    const int k0 = 4 * kc + 2 * hi;
#pragma unroll
    for (int nt = 0; nt < 4; ++nt) {
      const int col = 16 * nt + lo;
      bf[kc][nt].x = A[(k0 + 0) * HS + col];
      bf[kc][nt].y = A[(k0 + 1) * HS + col];
    }
  }

  // ---- t = 0 : u0 = E0 * I ; alpha = u0/Z ; ll = log(Z) ----
  v2f af[16];
  float ll;
  {
    const float* e = Bm + (size_t)obs_row[0] * HS;
    float s = 0.f;
#pragma unroll
    for (int kc = 0; kc < 16; ++kc) {
      const int k0 = 4 * kc + 2 * hi;
      float u0 = e[k0 + 0] * I[k0 + 0];
      float u1 = e[k0 + 1] * I[k0 + 1];
      af[kc].x = u0;
      af[kc].y = u1;
      s += u0 + u1;
    }
    float Z = s + __shfl_xor(s, 16, 32);
    ll = logf(Z);
    float inv = 1.0f / Z;
#pragma unroll
    for (int kc = 0; kc < 16; ++kc) {
      af[kc].x *= inv;
      af[kc].y *= inv;
    }
  }

  // ---- t = 1 .. T-1 ----
  for (int t = 1; t < HT; ++t) {
    const float* e = Bm + (size_t)obs_row[t] * HS;   // emission row (L0/L2 hot)

    // R = alpha @ A : 16 K-chunks x 4 N-tiles, fp32 WMMA accumulation
    v8f d[4] = {};
#pragma unroll
    for (int kc = 0; kc < 16; ++kc) {
#pragma unroll
      for (int nt = 0; nt < 4; ++nt) {
        d[nt] = __builtin_amdgcn_wmma_f32_16x16x4_f32(
            false, af[kc], false, bf[kc][nt], (short)0, d[nt], false, false);
      }
    }

    // D-layout -> LDS, column-major [col s][row m], stride 19
#pragma unroll
    for (int nt = 0; nt < 4; ++nt) {
      const int base = (16 * nt + lo) * LSTR + 8 * hi;
#pragma unroll
      for (int r = 0; r < 8; ++r) lds[base + r] = d[nt][r];
    }
    __syncthreads();   // single-wave WG: lowers to NOP, acts as LDS fence

    // Rebuild A-operand frags fused with emission mult; accumulate row sum
    float s = 0.f;
#pragma unroll
    for (int kc = 0; kc < 16; ++kc) {
      const int k0 = 4 * kc + 2 * hi;
      float u0 = lds[(k0 + 0) * LSTR + lo] * e[k0 + 0];
      float u1 = lds[(k0 + 1) * LSTR + lo] * e[k0 + 1];
      af[kc].x = u0;
      af[kc].y = u1;
      s += u0 + u1;
    }
    __syncthreads();   // WAR fence before next iteration's LDS stores

    float Z = s + __shfl_xor(s, 16, 32);   // both half-waves get full row sum
    ll += logf(Z);
    float inv = 1.0f / Z;
#pragma unroll
    for (int kc = 0; kc < 16; ++kc) {
      af[kc].x *= inv;
      af[kc].y *= inv;
    }
  }

  if (lane < 16) out[m] = ll;
}

// ---------------------------------------------------------------------------
extern "C" void kernel_launch(void* const* d_in, const int* in_sizes, int n_in,
                              void* d_out, int out_size, void* d_ws, size_t ws_size,
                              hipStream_t stream) {
  const float* x  = (const float*)d_in[0];   // [B, T, M] one-hot fp32
  const float* I  = (const float*)d_in[1];   // [1, S]
  const float* A  = (const float*)d_in[2];   // [S, S]
  const float* Bm = (const float*)d_in[3];   // [M, S]
  float* out = (float*)d_out;                // [B, 1]
  int* obs = (int*)d_ws;                     // [B*T] scratch (8 MB)

  (void)in_sizes; (void)n_in; (void)out_size; (void)ws_size;

  const int nrows = HB * HT;                 // 2,097,152 rows
  const int blocks1 = (nrows * 32 + 255) / 256;
  obs_from_onehot<<<blocks1, 256, 0, stream>>>(x, obs, nrows);

  hmm_forward_scan<<<HB / 16, 32, 0, stream>>>(obs, I, A, Bm, out);
}